// PoseModule_71227737637324
// MI455X (gfx1250) — compile-verified
//
#include <hip/hip_runtime.h>
#include <math.h>

// ---------------------------------------------------------------------------
// PoseModule for MI455X (gfx1250, wave32).
// Output-bandwidth-bound (671 MB out @ 23.3 TB/s ~ 29 us floor).
// The 3x3 rotation GEMM is mapped onto V_WMMA_F32_16X16X4_F32 by packing
// 5 poses (15 rows) into the 16-row A operand; B holds a 16-point grid tile.
// All guards are branchless (cndmask on store address into a d_ws dump) so
// the inner loop is pure load/WMMA/wrap/NT-store with EXEC always all-ones.
// ---------------------------------------------------------------------------

typedef __attribute__((ext_vector_type(2))) float v2f;
typedef __attribute__((ext_vector_type(4))) float v4f;
typedef __attribute__((ext_vector_type(8))) float v8f;

#define RES   128
#define NPTS  (RES * RES)            // 16384 grid points
#define BATCH 2048
#define GROUP 5                      // poses per WMMA (5*3 = 15 of 16 A rows)
#define NGROUPS ((BATCH + GROUP - 1) / GROUP)          // 410
#define TILES_PER_WAVE 8
#define WAVES_PER_GROUP (NPTS / 16 / TILES_PER_WAVE)   // 1024/8 = 128
#define PTS_OUT_ELEMS ((size_t)BATCH * 3 * NPTS)       // 100,663,296 floats

__device__ __forceinline__ float wrap_pi(float x) {
    // jnp.mod(x + pi, 2pi) - pi  (floor-mod)
    const float INV2PI = 0.15915494309189535f;
    const float TWOPI  = 6.283185307179586f;
    const float PI     = 3.14159265358979323846f;
    float t = (x + PI) * INV2PI;
    t -= floorf(t);
    return t * TWOPI - PI;
}

// ---------------------------------------------------------------------------
// pts_rot kernel: one wave owns (pose-group of 5, 8 tiles of 16 grid points).
// A (16x4 f32, ISA layout: lanes 0-15 hold K0/K1, lanes 16-31 hold K2/K3,
// M = lane%16): row m = 3*bi + r carries R[pose bi] row r.  B (4x16): rows
// 0..2 = grid x/y/z for 16 consecutive points.  D: lane = column (grid pt),
// VGPR v = row v + 8*(lane/16).
// ---------------------------------------------------------------------------
__global__ void __launch_bounds__(256) pts_wmma_kernel(
    const int*   __restrict__ index,
    const float* __restrict__ rotvec_weight,
    const float* __restrict__ grid,   // 3 x NPTS (x row, y row, z row)
    float*       __restrict__ out,    // BATCH x 3 x NPTS
    float*       __restrict__ dump)   // d_ws: sink for invalid-lane stores
{
    const int lane = threadIdx.x & 31;
    const int w    = blockIdx.x * 8 + (threadIdx.x >> 5);  // global wave id
    const int group     = w / WAVES_PER_GROUP;             // 0..409
    const int tile_base = (w % WAVES_PER_GROUP) * TILES_PER_WAVE;

    const int m    = lane & 15;   // A row index / D column index
    const int half = lane >> 4;   // 0: K=0,1   1: K=2,3

    // ---- Build A: Rodrigues per lane, branchless (clamped gather) ----------
    {
        // placeholder scope; A built below
    }
    const int  mi    = (m < 15) ? m : 0;        // clamp row for computation
    const int  bi_a  = mi / 3;
    const int  r_a   = mi - 3 * bi_a;
    int        b_a   = group * GROUP + bi_a;
    b_a = (b_a > BATCH - 1) ? (BATCH - 1) : b_a;
    const int  idx_a = index[b_a];
    const float rx = rotvec_weight[idx_a * 3 + 0];
    const float ry = rotvec_weight[idx_a * 3 + 1];
    const float rz = rotvec_weight[idx_a * 3 + 2];
    const float th  = sqrtf(rx * rx + ry * ry + rz * rz);
    const float inv = 1.0f / (th + 1e-6f);
    const float kx = rx * inv, ky = ry * inv, kz = rz * inv;
    const float s  = __sinf(th);
    const float c1 = 1.0f - __cosf(th);
    const float kk = kx * kx + ky * ky + kz * kz;   // K^2 = k k^T - kk*I
    // Row r_a of skew(k)
    const float K0 = (r_a == 0) ? 0.0f : ((r_a == 1) ?  kz  : -ky);
    const float K1 = (r_a == 0) ? -kz  : ((r_a == 1) ? 0.0f :  kx);
    const float K2 = (r_a == 0) ?  ky  : ((r_a == 1) ? -kx  : 0.0f);
    const float kr = (r_a == 0) ?  kx  : ((r_a == 1) ?  ky  :  kz);
    const float d0 = (r_a == 0) ? 1.0f : 0.0f;
    const float d1 = (r_a == 1) ? 1.0f : 0.0f;
    const float d2 = (r_a == 2) ? 1.0f : 0.0f;
    const float R0 = d0 * (1.0f - c1 * kk) + s * K0 + c1 * kr * kx;
    const float R1 = d1 * (1.0f - c1 * kk) + s * K1 + c1 * kr * ky;
    const float R2 = d2 * (1.0f - c1 * kk) + s * K2 + c1 * kr * kz;
    v2f a;
    a.x = (m < 15) ? (half ? R2   : R0) : 0.0f;   // K = 2*half
    a.y = (m < 15) ? (half ? 0.0f : R1) : 0.0f;   // K = 2*half+1 (K=3 pad)

    // ---- Precompute 8 branchless store base pointers (one per D VGPR) -----
    // Row mm = v + 8*half -> pose bi = mm/3, R-row r = mm%3, output row 2-r.
    // Invalid lanes (mm==15 or pose beyond BATCH) write into dump (d_ws).
    float* ptrs[8];
    #pragma unroll
    for (int v = 0; v < 8; ++v) {
        int mm = v + 8 * half;
        int bi = mm / 3;
        int r  = mm - 3 * bi;
        int b  = group * GROUP + bi;
        bool valid = (mm < 15) && (b < BATCH);
        size_t off = ((size_t)b * 3 + (size_t)(2 - r)) * NPTS
                   + (size_t)(tile_base * 16 + m);
        ptrs[v] = valid ? (out + off) : (dump + (size_t)(threadIdx.x));
    }

    // ---- 8 tiles of 16 grid points each (EXEC all-ones throughout) --------
    for (int t = 0; t < TILES_PER_WAVE; ++t) {
        const int p = (tile_base + t) * 16 + m;   // this lane's grid point

        // Load all three grid rows (L2/WGP$-resident), select branchlessly.
        const float g0 = grid[p];                 // x row
        const float g1 = grid[NPTS + p];          // y row
        const float g2 = grid[2 * NPTS + p];      // z row
        v2f bv;
        bv.x = half ? g2   : g0;
        bv.y = half ? 0.0f : g1;

        v8f c = {};
        c = __builtin_amdgcn_wmma_f32_16x16x4_f32(
                /*neg_a=*/false, a, /*neg_b=*/false, bv,
                /*c_mod=*/(short)0, c, /*reuse_a=*/false, /*reuse_b=*/false);

        #pragma unroll
        for (int v = 0; v < 8; ++v) {
            float val = wrap_pi(c[v]);
            __builtin_nontemporal_store(val, ptrs[v] + (size_t)t * 16);
        }
    }
}

// ---------------------------------------------------------------------------
// phase kernel: elementwise exp(-i*(gx*off0 + gy*off1)); 2 complex pts/thread
// -> one 16B non-temporal store. Pure store-bandwidth.
// ---------------------------------------------------------------------------
__global__ void __launch_bounds__(256) phase_kernel(
    const int*   __restrict__ index,
    const float* __restrict__ offsets_weight,
    const float* __restrict__ gx,
    const float* __restrict__ gy,
    float*       __restrict__ out)   // BATCH*NPTS complex64 (interleaved)
{
    const int blocks_per_b = NPTS / (256 * 2);          // 32
    const int b     = blockIdx.x / blocks_per_b;
    const int chunk = blockIdx.x % blocks_per_b;
    const int p     = chunk * 512 + threadIdx.x * 2;

    const int idx = index[b];
    const float o0 = -offsets_weight[idx * 2 + 0];
    const float o1 = -offsets_weight[idx * 2 + 1];

    v2f x2 = *(const v2f*)(gx + p);
    v2f y2 = *(const v2f*)(gy + p);
    float a0 = x2.x * o0 + y2.x * o1;
    float a1 = x2.y * o0 + y2.y * o1;

    v4f r;
    r.x = __cosf(a0); r.y = __sinf(a0);
    r.z = __cosf(a1); r.w = __sinf(a1);
    __builtin_nontemporal_store(r, (v4f*)(out + (size_t)b * NPTS * 2 + (size_t)p * 2));
}

extern "C" void kernel_launch(void* const* d_in, const int* in_sizes, int n_in,
                              void* d_out, int out_size, void* d_ws, size_t ws_size,
                              hipStream_t stream) {
    (void)in_sizes; (void)n_in; (void)out_size; (void)ws_size;
    const int*   index  = (const int*)  d_in[0];   // (2048,) int32
    const float* rotvec = (const float*)d_in[1];   // (100000,3)
    const float* offs   = (const float*)d_in[2];   // (100000,2)
    const float* grid   = (const float*)d_in[3];   // (3,16384)
    const float* gx     = (const float*)d_in[4];   // (128,128)
    const float* gy     = (const float*)d_in[5];   // (128,128)
    float* out  = (float*)d_out;
    float* dump = (float*)d_ws;                    // garbage sink (scratch)

    // pts_rot: 410 groups * 128 waves/group = 52480 waves, 8 waves/block
    const int pts_blocks = (NGROUPS * WAVES_PER_GROUP) / 8;       // 6560
    pts_wmma_kernel<<<pts_blocks, 256, 0, stream>>>(index, rotvec, grid, out, dump);

    // phase: 2048 poses * 32 blocks each
    const int ph_blocks = BATCH * (NPTS / 512);                   // 65536
    phase_kernel<<<ph_blocks, 256, 0, stream>>>(index, offs, gx, gy,
                                                out + PTS_OUT_ELEMS);
}